// RN_ComputeLoss_21912923144264
// MI455X (gfx1250) — compile-verified
//
#include <hip/hip_runtime.h>
#include <math.h>

typedef __attribute__((ext_vector_type(2))) float v2f;
typedef __attribute__((ext_vector_type(8))) float v8f;

#define NC        80
#define NBINS     16
#define CLIP_MAX  14.99f     // REG_MAX - 0.01
#define EPS       1e-7f
#define INV_PI2_4 0.4052847345693511f   // 4 / pi^2
#define BOX_W     7.5f
#define CLS_W     0.5f
#define DFL_W     1.5f

// focal BCE term for one (logit, target) pair.
// Identities used (exact in reals):
//   bce  = t*sp(-x) + (1-t)*sp(x) = sp(x) - t*x,  sp(x) = max(x,0) + log(1+e^-|x|)
//   p    = sigmoid(x) = (x>=0 ? 1 : e) / (1+e),   e = exp(-|x|)
//   p_t  = (1-p) + t*(2p-1);  alpha_f = 0.75 - 0.5*t;  mod = (1-p_t)^1.5
// -> one v_exp, one v_log, one v_rcp, one v_sqrt per class.
__device__ __forceinline__ float focal_term(float x, float t) {
    float ax  = fabsf(x);
    float e   = __expf(-ax);
    float L   = __logf(1.f + e);
    float bce = fmaxf(x, 0.f) + L - t * x;
    float r   = __builtin_amdgcn_rcpf(1.f + e);
    float p   = ((x >= 0.f) ? 1.f : e) * r;
    float pt  = (1.f - p) + t * (2.f * p - 1.f);
    float u   = 1.f - pt;                   // in (0,1)
    float af  = 0.75f - 0.5f * t;
    return bce * af * u * __builtin_amdgcn_sqrtf(u);
}

// one DFL side: softmax expectation (dist) + left/right weighted CE (dfl)
// 16 logits kept in registers; bin selection via compare-select (no dyn index)
__device__ __forceinline__ void dfl_side(const float4* __restrict__ q4, float tgt,
                                         float& dist, float& dfl) {
    float l[16];
    float4 q;
    q = q4[0]; l[0]=q.x;  l[1]=q.y;  l[2]=q.z;  l[3]=q.w;
    q = q4[1]; l[4]=q.x;  l[5]=q.y;  l[6]=q.z;  l[7]=q.w;
    q = q4[2]; l[8]=q.x;  l[9]=q.y;  l[10]=q.z; l[11]=q.w;
    q = q4[3]; l[12]=q.x; l[13]=q.y; l[14]=q.z; l[15]=q.w;
    float m = l[0];
#pragma unroll
    for (int k = 1; k < 16; ++k) m = fmaxf(m, l[k]);
    float se = 0.f, sk = 0.f;
#pragma unroll
    for (int k = 0; k < 16; ++k) {
        float e = __expf(l[k] - m);
        se += e; sk += (float)k * e;
    }
    dist = sk / se;
    float logZ = m + __logf(se);
    int   tli  = (int)tgt;                    // tgt >= 0 -> trunc == floor
    float wl   = (float)(tli + 1) - tgt;
    float wr   = 1.f - wl;
    float ll = 0.f, lr = 0.f;
#pragma unroll
    for (int k = 0; k < 16; ++k) {
        ll = (k == tli)     ? l[k] : ll;
        lr = (k == tli + 1) ? l[k] : lr;
    }
    dfl = (logZ - ll) * wl + (logZ - lr) * wr;
}

// acc layout: [0]=cls, [1]=iou, [2]=dfl, [3]=target_scores_sum
__global__ __launch_bounds__(256)
void yolo_loss_main(const float* __restrict__ pred_dist,
                    const float* __restrict__ pred_scores,
                    const float* __restrict__ anchor_points,
                    const float* __restrict__ target_bboxes,
                    const float* __restrict__ target_scores,
                    const unsigned char* __restrict__ fg_mask,
                    float* __restrict__ acc,
                    int A, int total) {
    int idx = blockIdx.x * blockDim.x + threadIdx.x;
    float s_cls = 0.f, s_iou = 0.f, s_dfl = 0.f, s_ts = 0.f;

    if (idx < total) {
        // ---- focal classification loss over all 80 classes (every anchor) ----
        const float4* ps4 = reinterpret_cast<const float4*>(pred_scores   + (size_t)idx * NC);
        const float4* ts4 = reinterpret_cast<const float4*>(target_scores + (size_t)idx * NC);
#pragma unroll
        for (int i = 0; i < NC / 4; ++i) {
            float4 x4 = ps4[i], t4 = ts4[i];
            s_cls += focal_term(x4.x, t4.x); s_ts += t4.x;
            s_cls += focal_term(x4.y, t4.y); s_ts += t4.y;
            s_cls += focal_term(x4.z, t4.z); s_ts += t4.z;
            s_cls += focal_term(x4.w, t4.w); s_ts += t4.w;
        }

        // ---- fg-only: skip the 256B pred_dist read for ~95% of anchors ----
        if (fg_mask[idx]) {
            int    a  = idx % A;
            float2 ap = reinterpret_cast<const float2*>(anchor_points)[a];
            float4 tb = reinterpret_cast<const float4*>(target_bboxes)[idx];

            float tgl = fminf(fmaxf(ap.x - tb.x, 0.f), CLIP_MAX);
            float tgt = fminf(fmaxf(ap.y - tb.y, 0.f), CLIP_MAX);
            float tgr = fminf(fmaxf(tb.z - ap.x, 0.f), CLIP_MAX);
            float tgb = fminf(fmaxf(tb.w - ap.y, 0.f), CLIP_MAX);

            const float4* dp4 = reinterpret_cast<const float4*>(pred_dist + (size_t)idx * 4 * NBINS);
            float d0, d1, d2, d3, f0, f1, f2, f3;
            dfl_side(dp4 + 0,  tgl, d0, f0);
            dfl_side(dp4 + 4,  tgt, d1, f1);
            dfl_side(dp4 + 8,  tgr, d2, f2);
            dfl_side(dp4 + 12, tgb, d3, f3);

            float px1 = ap.x - d0, py1 = ap.y - d1;
            float px2 = ap.x + d2, py2 = ap.y + d3;

            // CIoU (forward value; stop_gradient irrelevant here)
            float w1 = px2 - px1,   h1 = py2 - py1 + EPS;
            float w2 = tb.z - tb.x, h2 = tb.w - tb.y + EPS;
            float iw = fmaxf(fminf(px2, tb.z) - fmaxf(px1, tb.x), 0.f);
            float ih = fmaxf(fminf(py2, tb.w) - fmaxf(py1, tb.y), 0.f);
            float inter = iw * ih;
            float uni   = w1 * h1 + w2 * h2 - inter + EPS;
            float iou   = inter / uni;
            float cw = fmaxf(px2, tb.z) - fminf(px1, tb.x);
            float ch = fmaxf(py2, tb.w) - fminf(py1, tb.y);
            float c2 = cw * cw + ch * ch + EPS;
            float dx = tb.x + tb.z - px1 - px2;
            float dy = tb.y + tb.w - py1 - py2;
            float rho2 = (dx * dx + dy * dy) * 0.25f;
            float dat  = atanf(w2 / h2) - atanf(w1 / h1);
            float v    = INV_PI2_4 * dat * dat;
            float alph = v / (v - iou + (1.f + EPS));
            float ciou = iou - (rho2 / c2 + v * alph);

            float w = s_ts;                 // bbox_weight (fg==1 in this branch)
            s_iou = (1.f - ciou) * w;
            s_dfl = ((f0 + f1 + f2 + f3) * 0.25f) * w;
        }
    }

    // ---- intra-wave (wave32) butterfly reduction ----
#pragma unroll
    for (int off = 16; off > 0; off >>= 1) {
        s_cls += __shfl_xor(s_cls, off, 32);
        s_iou += __shfl_xor(s_iou, off, 32);
        s_dfl += __shfl_xor(s_dfl, off, 32);
        s_ts  += __shfl_xor(s_ts,  off, 32);
    }

    // 8 waves x 16 columns, zero-padded so WMMA B-feed loads are branch-free
    __shared__ float wpart[8 * 16];
    int wave = threadIdx.x >> 5;
    int lane = threadIdx.x & 31;
    if (lane < 16) {
        float v = (lane == 0) ? s_cls :
                  (lane == 1) ? s_iou :
                  (lane == 2) ? s_dfl :
                  (lane == 3) ? s_ts  : 0.f;
        wpart[wave * 16 + lane] = v;
    }
    __syncthreads();

    // ---- cross-wave reduction via two chained f32 WMMAs (exact precision) ----
    // D = ones(16x4) x B(4x16) + C sums each B column over K.
    // B layout (32-bit, 4x16): lanes 0-15 -> N, VGPR0=K0/VGPR1=K1;
    //                          lanes 16-31 -> N, VGPR0=K2/VGPR1=K3.
    // Column n<4 carries quantity n of 4 waves per pass; pass 2 accumulates via C.
    if (threadIdx.x < 32) {                 // whole wave 0: EXEC all ones
        int n = lane & 15, half = lane >> 4;
        v2f aa; aa.x = 1.f; aa.y = 1.f;     // ones A-matrix
        v8f c = {};
        v2f b;
        b.x = wpart[(2 * half + 0) * 16 + n];          // waves 0..3
        b.y = wpart[(2 * half + 1) * 16 + n];
        c = __builtin_amdgcn_wmma_f32_16x16x4_f32(false, aa, false, b,
                                                  (short)0, c, false, false);
        b.x = wpart[(4 + 2 * half + 0) * 16 + n];      // waves 4..7
        b.y = wpart[(4 + 2 * half + 1) * 16 + n];
        c = __builtin_amdgcn_wmma_f32_16x16x4_f32(false, aa, false, b,
                                                  (short)0, c, false, false);
        if (lane < 4) atomicAdd(&acc[lane], c[0]);     // D[0][n] = block total n
    }
}

__global__ void yolo_loss_finalize(const float* __restrict__ acc,
                                   float* __restrict__ out) {
    float tss = fmaxf(acc[3], 1.0f);
    out[0] = BOX_W * acc[1] / tss;   // box
    out[1] = CLS_W * acc[0] / tss;   // cls
    out[2] = DFL_W * acc[2] / tss;   // dfl
}

extern "C" void kernel_launch(void* const* d_in, const int* in_sizes, int n_in,
                              void* d_out, int out_size, void* d_ws, size_t ws_size,
                              hipStream_t stream) {
    const float*         pred_dist     = (const float*)d_in[0];
    const float*         pred_scores   = (const float*)d_in[1];
    const float*         anchor_points = (const float*)d_in[2];
    const float*         target_bboxes = (const float*)d_in[3];
    const float*         target_scores = (const float*)d_in[4];
    const unsigned char* fg_mask       = (const unsigned char*)d_in[5];

    int A     = in_sizes[2] / 2;     // 8400
    int total = in_sizes[5];         // B * A = 268800

    float* acc = (float*)d_ws;       // 4 f32 accumulators
    hipMemsetAsync(acc, 0, 4 * sizeof(float), stream);

    const int threads = 256;
    int blocks = (total + threads - 1) / threads;
    yolo_loss_main<<<blocks, threads, 0, stream>>>(
        pred_dist, pred_scores, anchor_points, target_bboxes,
        target_scores, fg_mask, acc, A, total);

    yolo_loss_finalize<<<1, 1, 0, stream>>>(acc, (float*)d_out);
}